// VolumeRotation_10900626997635
// MI455X (gfx1250) — compile-verified
//
#include <hip/hip_runtime.h>
#include <hip/hip_bf16.h>

typedef __attribute__((ext_vector_type(2))) float v2f;
typedef __attribute__((ext_vector_type(8))) float v8f;

#define SZ   64
#define SSS  (SZ * SZ * SZ)
#define NCH  16

__global__ __launch_bounds__(256)
void VolumeRotation_kernel(const float* __restrict__ vol,
                           const float* __restrict__ rot,
                           float* __restrict__ out)
{
    const int tid  = threadIdx.x;
    const int lane = tid & 31;                    // wave32
    const bool lo  = lane < 16;

    // Wave covers 32 consecutive x positions of one (b, z, y) row-half.
    const int wavebase = blockIdx.x * 256 + (tid & ~31);
    const int xw0 = wavebase & (SZ - 1);
    const int y   = (wavebase >> 6)  & (SZ - 1);
    const int z   = (wavebase >> 12) & (SZ - 1);
    const int b   =  wavebase >> 18;

    const float step = 2.0f / 63.0f;              // linspace(-1,1,64) step
    const float by = (float)y * step - 1.0f;
    const float bz = (float)z * step - 1.0f;

    // ---- A matrix (16x4 f32), lane l holds row l%16, K pair {0,1} (lanes 0-15)
    // or {2,3} (lanes 16-31).  Rows 0-2 and 8-10 = U^T (UT[i][j] = rot[j*3+i]);
    // other rows and column K=3 are zero.
    const int row = lane & 15;
    const int r8  = row & 7;
    float a0 = 0.0f, a1 = 0.0f;
    if (r8 < 3) {
        a0 = rot[b * 9 + (lo ? 0 : 6) + r8];      // K = 0 or 2
        if (lo) a1 = rot[b * 9 + 3 + r8];         // K = 1 (K=3 column is zero)
    }

    // ---- B matrices (4x16 f32): lane l holds column l%16, K pair split like A.
    // Rows: K0 = x-base, K1 = y-base, K2 = z-base, K3 = 0.
    const int col  = lane & 15;
    const float bx1 = (float)(xw0 + col)      * step - 1.0f;
    const float bx2 = (float)(xw0 + 16 + col) * step - 1.0f;
    v2f A  = { a0, a1 };
    v2f B1 = { lo ? bx1 : bz, lo ? by : 0.0f };
    v2f B2 = { lo ? bx2 : bz, lo ? by : 0.0f };
    v8f cz = {};
    // D = U^T * base  for 16 positions per WMMA (rows 0-2 / 8-10 hold gx,gy,gz)
    v8f D1 = __builtin_amdgcn_wmma_f32_16x16x4_f32(false, A, false, B1,
                                                   (short)0, cz, false, false);
    v8f D2 = __builtin_amdgcn_wmma_f32_16x16x4_f32(false, A, false, B2,
                                                   (short)0, cz, false, false);
    // lane<16 reads D rows 0-2 (cols = xw0+lane); lane>=16 reads rows 8-10 of D2
    // (cols = xw0+16+(lane-16) = xw0+lane) -> every lane gets its own coordinate.
    const float gx = lo ? D1[0] : D2[0];
    const float gy = lo ? D1[1] : D2[1];
    const float gz = lo ? D1[2] : D2[2];

    // ---- grid_sample: align_corners=False, padding_mode='zeros'
    const float xs = (gx + 1.0f) * 32.0f - 0.5f;
    const float ys = (gy + 1.0f) * 32.0f - 0.5f;
    const float zs = (gz + 1.0f) * 32.0f - 0.5f;
    const float fx = floorf(xs), fy = floorf(ys), fz = floorf(zs);
    const float tx = xs - fx,    ty = ys - fy,    tz = zs - fz;
    const int x0 = (int)fx, y0 = (int)fy, z0 = (int)fz;
    const int x1 = x0 + 1,  y1 = y0 + 1,  z1 = z0 + 1;

    // 1D weights with validity folded in (== reference's w * valid)
    const float wx0 = (1.0f - tx) * ((x0 >= 0 && x0 < SZ) ? 1.0f : 0.0f);
    const float wx1 =         tx  * ((x1 >= 0 && x1 < SZ) ? 1.0f : 0.0f);
    const float wy0 = (1.0f - ty) * ((y0 >= 0 && y0 < SZ) ? 1.0f : 0.0f);
    const float wy1 =         ty  * ((y1 >= 0 && y1 < SZ) ? 1.0f : 0.0f);
    const float wz0 = (1.0f - tz) * ((z0 >= 0 && z0 < SZ) ? 1.0f : 0.0f);
    const float wz1 =         tz  * ((z1 >= 0 && z1 < SZ) ? 1.0f : 0.0f);

    const float w000 = wx0 * wy0 * wz0, w100 = wx1 * wy0 * wz0;
    const float w010 = wx0 * wy1 * wz0, w110 = wx1 * wy1 * wz0;
    const float w001 = wx0 * wy0 * wz1, w101 = wx1 * wy0 * wz1;
    const float w011 = wx0 * wy1 * wz1, w111 = wx1 * wy1 * wz1;

    // clamped indices (invalid corners carry zero weight)
    const int xc0 = min(max(x0, 0), SZ - 1), xc1 = min(max(x1, 0), SZ - 1);
    const int yc0 = min(max(y0, 0), SZ - 1), yc1 = min(max(y1, 0), SZ - 1);
    const int zc0 = min(max(z0, 0), SZ - 1), zc1 = min(max(z1, 0), SZ - 1);

    const int r00 = (zc0 * SZ + yc0) * SZ, r10 = (zc0 * SZ + yc1) * SZ;
    const int r01 = (zc1 * SZ + yc0) * SZ, r11 = (zc1 * SZ + yc1) * SZ;
    const int i000 = r00 + xc0, i100 = r00 + xc1;
    const int i010 = r10 + xc0, i110 = r10 + xc1;
    const int i001 = r01 + xc0, i101 = r01 + xc1;
    const int i011 = r11 + xc0, i111 = r11 + xc1;

    const int    outlin = (z * SZ + y) * SZ + (xw0 + lane);
    const float* vp = vol + (size_t)(b * NCH) * SSS;
    float*       op = out + (size_t)(b * NCH) * SSS + outlin;

    #pragma unroll
    for (int ch = 0; ch < NCH; ++ch) {
        const float* p = vp + ch * SSS;
        float acc = w000 * p[i000] + w100 * p[i100]
                  + w010 * p[i010] + w110 * p[i110]
                  + w001 * p[i001] + w101 * p[i101]
                  + w011 * p[i011] + w111 * p[i111];
        op[ch * SSS] = acc;
    }
}

extern "C" void kernel_launch(void* const* d_in, const int* in_sizes, int n_in,
                              void* d_out, int out_size, void* d_ws, size_t ws_size,
                              hipStream_t stream) {
    const float* vol = (const float*)d_in[0];   // [8,16,64,64,64] f32
    const float* rot = (const float*)d_in[1];   // [8,3,3] f32
    float*       out = (float*)d_out;           // [8,16,64,64,64] f32

    const int positions = 8 * SSS;              // 2,097,152 -> 8192 full blocks
    dim3 grid(positions / 256), block(256);
    hipLaunchKernelGGL(VolumeRotation_kernel, grid, block, 0, stream,
                       vol, rot, out);
}